// FastGrokkingRushNetWithGAT_13365938225773
// MI455X (gfx1250) — compile-verified
//
#include <hip/hip_runtime.h>

// ---------------------------------------------------------------------------
// Types for CDNA5 WMMA f32 16x16x4:  A is 16x4 f32 (2 VGPRs/lane),
// B is 4x16 f32 (2 VGPRs/lane), C/D is 16x16 f32 (8 VGPRs/lane).
// ---------------------------------------------------------------------------
typedef __attribute__((ext_vector_type(2))) float v2f;
typedef __attribute__((ext_vector_type(8))) float v8f;

#define HEADS 4
#define DHEAD 64
#define HDIM 256   // HEADS*DHEAD == HID == all GEMM output widths

// ---------------------------------------------------------------------------
// f32 WMMA GEMM, Nc fixed at 256:  C[M,256] = act( (A + abias_k) @ B + obias )
//   - block = 256 threads = 8 waves; each wave owns a 32x32 output tile:
//     2 A-fragments x 2 B-fragments -> 4 WMMAs per K-step (1.5 loads/WMMA)
//   - grid.x = ceil(M/32); M%16==0. If the second 16-row tile would overrun,
//     it is clamped to M-16 (in-bounds loads, redundant compute) and its
//     stores are suppressed by a wave-uniform flag -> EXEC stays all-ones
//     through every WMMA (ISA requirement), no divergence.
//   - ABIAS/OBIAS/RELU are compile-time so the inner loop stays branch-free
// A-fragment (16x4): lane m=lane&15 holds row m; lanes>=16 hold K+2,K+3.
// B-fragment (4x16): lane n=lane&15 holds col n; lanes>=16 hold K+2,K+3.
// C/D: VGPR r -> M=r (lanes 0-15) / M=r+8 (lanes 16-31), N=lane&15.
// ---------------------------------------------------------------------------
template<bool ABIAS, bool OBIAS, bool RELU>
__global__ void __launch_bounds__(256)
wmma_gemm_f32_kernel(const float* __restrict__ A, const float* __restrict__ B,
                     const float* __restrict__ abias,   // k-indexed, added to A rows
                     const float* __restrict__ obias,   // column bias on output
                     float* __restrict__ C, int M, int K) {
    constexpr int NC = 256;
    const int lane = threadIdx.x & 31;
    const int wave = threadIdx.x >> 5;
    const int rowA = blockIdx.x * 32;
    int rowB = rowA + 16;
    const bool doRow1 = (rowB + 16 <= M);    // wave-uniform tail guard
    if (!doRow1) rowB = M - 16;              // clamp: loads stay in-bounds
    const int col0 = wave * 32;              // 8 waves * 32 cols = 256
    const int m    = lane & 15;
    const int n    = lane & 15;
    const int koff = (lane >> 4) << 1;       // 0 for lanes 0-15, 2 for lanes 16-31

    // running pointers -> no per-iteration 64-bit multiplies
    const float* ap0 = A + (size_t)(rowA + m) * K + koff;
    const float* ap1 = A + (size_t)(rowB + m) * K + koff;
    const float* bp  = B + koff * NC + col0 + n;
    const float* abp = ABIAS ? (abias + koff) : nullptr;

    v8f acc00 = {}, acc01 = {}, acc10 = {}, acc11 = {};

#pragma unroll 4
    for (int k = 0; k < K; k += 4) {
        v2f a0 = *(const v2f*)ap0;           // 8B aligned (koff even, K%4==0)
        v2f a1 = *(const v2f*)ap1;
        if (ABIAS) {
            const float ab0 = abp[0], ab1 = abp[1];
            a0.x += ab0; a0.y += ab1;
            a1.x += ab0; a1.y += ab1;
            abp += 4;
        }
        v2f b0, b1;
        b0.x = bp[0];        b0.y = bp[NC];          // imm offsets 0 / 0x400
        b1.x = bp[16];       b1.y = bp[NC + 16];     // imm offsets 0x40 / 0x440
        acc00 = __builtin_amdgcn_wmma_f32_16x16x4_f32(
                    false, a0, false, b0, (short)0, acc00, false, false);
        acc01 = __builtin_amdgcn_wmma_f32_16x16x4_f32(
                    false, a0, false, b1, (short)0, acc01, false, false);
        acc10 = __builtin_amdgcn_wmma_f32_16x16x4_f32(
                    false, a1, false, b0, (short)0, acc10, false, false);
        acc11 = __builtin_amdgcn_wmma_f32_16x16x4_f32(
                    false, a1, false, b1, (short)0, acc11, false, false);
        ap0 += 4;
        ap1 += 4;
        bp  += 4 * NC;
    }

    const float ob0 = OBIAS ? obias[col0 + n]      : 0.0f;
    const float ob1 = OBIAS ? obias[col0 + 16 + n] : 0.0f;
    const int rbase = (lane >> 4) * 8;
#pragma unroll
    for (int r = 0; r < 8; ++r) {
        float v00 = acc00[r] + ob0;
        float v01 = acc01[r] + ob1;
        if (RELU) { v00 = v00 > 0.0f ? v00 : 0.0f; v01 = v01 > 0.0f ? v01 : 0.0f; }
        float* crowA = C + (size_t)(rowA + rbase + r) * NC + col0 + n;
        crowA[0]  = v00;
        crowA[16] = v01;
    }
    if (doRow1) {
#pragma unroll
        for (int r = 0; r < 8; ++r) {
            float v10 = acc10[r] + ob0;
            float v11 = acc11[r] + ob1;
            if (RELU) { v10 = v10 > 0.0f ? v10 : 0.0f; v11 = v11 > 0.0f ? v11 : 0.0f; }
            float* crowB = C + (size_t)(rowB + rbase + r) * NC + col0 + n;
            crowB[0]  = v10;
            crowB[16] = v11;
        }
    }
}

// ---------------------------------------------------------------------------
// Attention coefficients: a_s[n,h] = <h[n,h,:], att_src[h,:]>, same for a_d.
// One wave per node; lane l handles elements 2l,2l+1 of each 64-wide head.
// ---------------------------------------------------------------------------
__global__ void attn_coef_kernel(const float* __restrict__ hmat,
                                 const float* __restrict__ att_src,
                                 const float* __restrict__ att_dst,
                                 float* __restrict__ a_s, float* __restrict__ a_d, int N) {
    const int lane = threadIdx.x & 31;
    const int wave = threadIdx.x >> 5;
    const int node = blockIdx.x * (blockDim.x >> 5) + wave;
    if (node >= N) return;
    const float2* hrow = (const float2*)(hmat + (size_t)node * HDIM);
#pragma unroll
    for (int h = 0; h < HEADS; ++h) {
        const float2 hv = hrow[h * 32 + lane];
        const float2 as = ((const float2*)att_src)[h * 32 + lane];
        const float2 ad = ((const float2*)att_dst)[h * 32 + lane];
        float s = hv.x * as.x + hv.y * as.y;
        float d = hv.x * ad.x + hv.y * ad.y;
#pragma unroll
        for (int off = 16; off > 0; off >>= 1) {
            s += __shfl_xor(s, off, 32);
            d += __shfl_xor(d, off, 32);
        }
        if (lane == 0) {
            a_s[node * HEADS + h] = s;
            a_d[node * HEADS + h] = d;
        }
    }
}

// Order-preserving float<->uint key for atomicMax on signed floats
__device__ __forceinline__ unsigned fkey(float x) {
    unsigned u = __float_as_uint(x);
    return (u & 0x80000000u) ? ~u : (u | 0x80000000u);
}
__device__ __forceinline__ float funkey(unsigned k) {
    unsigned u = (k & 0x80000000u) ? (k & 0x7fffffffu) : ~k;
    return __uint_as_float(u);
}
__device__ __forceinline__ float leaky(float x) { return x > 0.0f ? x : 0.2f * x; }

// Pass 1: per-(dst,head) max of leaky-relu logits. One thread per edge (incl. self-loops).
__global__ void edge_max_kernel(const long long* __restrict__ ei,
                                const float* __restrict__ a_s, const float* __restrict__ a_d,
                                unsigned* __restrict__ emax, int E, int N) {
    const long long i = (long long)blockIdx.x * blockDim.x + threadIdx.x;
    if (i >= (long long)E + N) return;
    long long s, t;
    if (i < E) { s = ei[i]; t = ei[E + i]; } else { s = t = i - E; }
#pragma unroll
    for (int h = 0; h < HEADS; ++h) {
        const float e = leaky(a_s[s * HEADS + h] + a_d[t * HEADS + h]);
        atomicMax(&emax[t * HEADS + h], fkey(e));
    }
}

// Pass 2: denom[dst,h] += exp(e - emax)
__global__ void edge_denom_kernel(const long long* __restrict__ ei,
                                  const float* __restrict__ a_s, const float* __restrict__ a_d,
                                  const unsigned* __restrict__ emax,
                                  float* __restrict__ denom, int E, int N) {
    const long long i = (long long)blockIdx.x * blockDim.x + threadIdx.x;
    if (i >= (long long)E + N) return;
    long long s, t;
    if (i < E) { s = ei[i]; t = ei[E + i]; } else { s = t = i - E; }
#pragma unroll
    for (int h = 0; h < HEADS; ++h) {
        const float e = leaky(a_s[s * HEADS + h] + a_d[t * HEADS + h]);
        const float w = __expf(e - funkey(emax[t * HEADS + h]));
        unsafeAtomicAdd(&denom[t * HEADS + h], w);
    }
}

// Pass 3: out[dst,h,:] += alpha * h[src,h,:]. Thread = (edge, head, d/4): float4 gather + 4 f32 atomics.
__global__ void edge_agg_kernel(const long long* __restrict__ ei,
                                const float* __restrict__ hmat,
                                const float* __restrict__ a_s, const float* __restrict__ a_d,
                                const unsigned* __restrict__ emax, const float* __restrict__ denom,
                                float* __restrict__ out, int E, int N) {
    const long long tid = (long long)blockIdx.x * blockDim.x + threadIdx.x;
    const long long total = ((long long)E + N) * (HEADS * DHEAD / 4);
    if (tid >= total) return;
    const long long i = tid >> 6;             // 64 threads per edge
    const int rem = (int)(tid & 63);
    const int h   = rem >> 4;                 // head
    const int d4  = (rem & 15) << 2;          // d offset (x4)
    long long s, t;
    if (i < E) { s = ei[i]; t = ei[E + i]; } else { s = t = i - E; }

    const float e = leaky(a_s[s * HEADS + h] + a_d[t * HEADS + h]);
    const float alpha = __expf(e - funkey(emax[t * HEADS + h])) / denom[t * HEADS + h];

    const float4 hv = *(const float4*)(hmat + s * HDIM + h * DHEAD + d4);
    float* o = out + t * HDIM + h * DHEAD + d4;
    unsafeAtomicAdd(o + 0, hv.x * alpha);
    unsafeAtomicAdd(o + 1, hv.y * alpha);
    unsafeAtomicAdd(o + 2, hv.z * alpha);
    unsafeAtomicAdd(o + 3, hv.w * alpha);
}

// Final 256->2 projection: one wave per node, shuffle-reduced dual dot product.
__global__ void final_proj_kernel(const float* __restrict__ h2,
                                  const float* __restrict__ W3, const float* __restrict__ b3,
                                  float* __restrict__ out, int N) {
    const int lane = threadIdx.x & 31;
    const int wave = threadIdx.x >> 5;
    const int node = blockIdx.x * (blockDim.x >> 5) + wave;
    if (node >= N) return;
    const float* row = h2 + (size_t)node * HDIM;
    float s0 = 0.0f, s1 = 0.0f;
#pragma unroll
    for (int i = 0; i < HDIM / 32; ++i) {
        const int k = i * 32 + lane;
        const float v = row[k];
        s0 += v * W3[k * 2 + 0];
        s1 += v * W3[k * 2 + 1];
    }
#pragma unroll
    for (int off = 16; off > 0; off >>= 1) {
        s0 += __shfl_xor(s0, off, 32);
        s1 += __shfl_xor(s1, off, 32);
    }
    if (lane == 0) {
        out[node * 2 + 0] = s0 + b3[0];
        out[node * 2 + 1] = s1 + b3[1];
    }
}

// ---------------------------------------------------------------------------
// Launcher
// ---------------------------------------------------------------------------
extern "C" void kernel_launch(void* const* d_in, const int* in_sizes, int n_in,
                              void* d_out, int out_size, void* d_ws, size_t ws_size,
                              hipStream_t stream) {
    const float*     x       = (const float*)d_in[0];
    const long long* ei      = (const long long*)d_in[1];   // int64 [2,E]
    const float*     W_gat   = (const float*)d_in[2];
    const float*     att_src = (const float*)d_in[3];
    const float*     att_dst = (const float*)d_in[4];
    const float*     b_gat   = (const float*)d_in[5];
    const float*     W1      = (const float*)d_in[6];
    const float*     b1      = (const float*)d_in[7];
    const float*     W2      = (const float*)d_in[8];
    const float*     b2      = (const float*)d_in[9];
    const float*     W3      = (const float*)d_in[10];
    const float*     b3      = (const float*)d_in[11];
    float*           outp    = (float*)d_out;

    const int F_IN = 128;
    const int N = in_sizes[0] / F_IN;       // 50000 (divisible by 16)
    const int E = in_sizes[1] / 2;          // 800000
    const int Etot = E + N;

    // Workspace layout (~106 MB): two ping-pong NxHDIM buffers + small arrays
    float*    hbuf  = (float*)d_ws;                         // h = x@W_gat, later h1
    float*    gbuf  = hbuf + (size_t)N * HDIM;              // gat_out, later h2
    float*    a_s   = gbuf + (size_t)N * HDIM;              // [N,4]
    float*    a_d   = a_s + (size_t)N * HEADS;
    unsigned* emax  = (unsigned*)(a_d + (size_t)N * HEADS); // max-keys [N,4]
    float*    denom = (float*)(emax + (size_t)N * HEADS);   // [N,4]

    // zero-init accumulators (0x0 == 0.0f == "no key yet")
    hipMemsetAsync(gbuf,  0, (size_t)N * HDIM  * sizeof(float),    stream);
    hipMemsetAsync(emax,  0, (size_t)N * HEADS * sizeof(unsigned), stream);
    hipMemsetAsync(denom, 0, (size_t)N * HEADS * sizeof(float),    stream);

    const dim3 blk(256);
    const dim3 gemmGrid((N + 31) / 32);

    // 1) h = x @ W_gat   [N,128]@[128,256]
    wmma_gemm_f32_kernel<false, false, false><<<gemmGrid, blk, 0, stream>>>(
        x, W_gat, nullptr, nullptr, hbuf, N, F_IN);

    // 2) attention coefficients
    attn_coef_kernel<<<dim3((N + 7) / 8), blk, 0, stream>>>(hbuf, att_src, att_dst, a_s, a_d, N);

    // 3-5) edge softmax + aggregation
    edge_max_kernel<<<dim3((Etot + 255) / 256), blk, 0, stream>>>(ei, a_s, a_d, emax, E, N);
    edge_denom_kernel<<<dim3((Etot + 255) / 256), blk, 0, stream>>>(ei, a_s, a_d, emax, denom, E, N);
    const long long aggT = (long long)Etot * 64;
    edge_agg_kernel<<<dim3((unsigned)((aggT + 255) / 256)), blk, 0, stream>>>(
        ei, hbuf, a_s, a_d, emax, denom, gbuf, E, N);

    // 6) h1 = relu((gat_out + b_gat) @ W1 + b1)   -> hbuf
    wmma_gemm_f32_kernel<true, true, true><<<gemmGrid, blk, 0, stream>>>(
        gbuf, W1, b_gat, b1, hbuf, N, HDIM);

    // 7) h2 = relu(h1 @ W2 + b2)                  -> gbuf
    wmma_gemm_f32_kernel<false, true, true><<<gemmGrid, blk, 0, stream>>>(
        hbuf, W2, nullptr, b2, gbuf, N, HDIM);

    // 8) out = h2 @ W3 + b3
    final_proj_kernel<<<dim3((N + 7) / 8), blk, 0, stream>>>(gbuf, W3, b3, outp, N);
}